// WildCatPooling_58420145160604
// MI455X (gfx1250) — compile-verified
//
#include <hip/hip_runtime.h>

// WildCat pooling on MI455X (gfx1250).
// x: [32, 512, 64, 64] f32 -> out: [32, 512] f32
// out[n,c] = mean(top-410 of row) + 0.6 * mean(bottom-410 of row), row = 4096 elems.
// One 256-thread block per (n,c) row. Exact 4-pass radix select over ordered-uint keys.
// Data path: CDNA5 GLOBAL_LOAD_ASYNC_TO_LDS_B128 + s_wait_asynccnt (ASYNCcnt).

#define HW       4096
#define KSEL     410          // round(0.1 * 4096), Python banker's rounding -> 410
#define THREADS  256
#define PER_TH   (HW / THREADS)   // 16 floats per thread
#define NWAVES   (THREADS / 32)   // 8 wave32 per block
#define NSUB     4                // sub-banked histograms to reduce ds_add conflicts
#define ALPHA    0.6f

__device__ __forceinline__ unsigned f2k(float f) {
    // order-preserving float -> uint (uint compare == float compare; no NaNs in input)
    unsigned u = __float_as_uint(f);
    return (u & 0x80000000u) ? ~u : (u | 0x80000000u);
}
__device__ __forceinline__ float k2f(unsigned k) {
    unsigned u = (k & 0x80000000u) ? (k & 0x7FFFFFFFu) : ~k;
    return __uint_as_float(u);
}

__global__ __launch_bounds__(THREADS)
void wildcat_pool_kernel(const float* __restrict__ x, float* __restrict__ out) {
    __shared__ __align__(16) float data[HW];                 // 16 KB row staging
    __shared__ unsigned hist[NWAVES][NSUB][256];             // 32 KB sub-banked histograms
    __shared__ unsigned scan[256];                           // 1 KB suffix-scan
    __shared__ float    wsum[NWAVES];
    __shared__ unsigned s_bucket, s_remk;

    const int tid  = threadIdx.x;
    const int wave = tid >> 5;          // wave32
    const int lane = tid & 31;
    const int sub  = lane & (NSUB - 1);
    const long long row = blockIdx.x;
    const float* __restrict__ src = x + row * (long long)HW;

    // ---------------- Phase 1: async global -> LDS (CDNA5 path) ----------------
    // Thread t owns floats {q*1024 + t*4 .. +3} for q=0..3: each b128 instruction is
    // a fully coalesced 512B/wave transfer; the signed INST_OFFSET is added to BOTH
    // the LDS and global addresses (ISA 08 section 4.4), so one VGPR pair suffices.
    {
        unsigned           ldsoff = (unsigned)(unsigned long long)&data[tid * 4];
        unsigned long long gaddr  = (unsigned long long)(src + tid * 4);
        asm volatile(
            "global_load_async_to_lds_b128 %0, %1, off\n\t"
            "global_load_async_to_lds_b128 %0, %1, off offset:4096\n\t"
            "global_load_async_to_lds_b128 %0, %1, off offset:8192\n\t"
            "global_load_async_to_lds_b128 %0, %1, off offset:12288"
            :: "v"(ldsoff), "v"(gaddr) : "memory");
        asm volatile("s_wait_asynccnt 0" ::: "memory");
    }
    __syncthreads();

    // Pull this thread's 16 elements into registers (ds_load_b128 x4).
    float vals[PER_TH];
#pragma unroll
    for (int q = 0; q < 4; ++q) {
        float4 v = *(const float4*)&data[q * 1024 + tid * 4];
        vals[q * 4 + 0] = v.x; vals[q * 4 + 1] = v.y;
        vals[q * 4 + 2] = v.z; vals[q * 4 + 3] = v.w;
    }

    // ---------------- Phase 2: two exact radix selects + sums ----------------
    float sum_top = 0.f, sum_bot = 0.f;

#pragma unroll 1
    for (int side = 0; side < 2; ++side) {
        const unsigned flip = side ? 0xFFFFFFFFu : 0u;   // side 1: invert order -> bottom-k
        unsigned prefix = 0, pmask = 0, remk = KSEL;

#pragma unroll 1
        for (int shift = 24; shift >= 0; shift -= 8) {
            // zero histograms (8*4*256 = 8192 words, 32 per thread)
            for (int i = tid; i < NWAVES * NSUB * 256; i += THREADS)
                ((unsigned*)hist)[i] = 0u;
            __syncthreads();

            // histogram current byte among elements matching the chosen prefix
#pragma unroll
            for (int e = 0; e < PER_TH; ++e) {
                unsigned key = f2k(vals[e]) ^ flip;
                if ((key & pmask) == prefix)
                    atomicAdd(&hist[wave][sub][(key >> shift) & 255u], 1u);
            }
            __syncthreads();

            // fold wave/sub copies, bucket `tid`
            unsigned c = 0;
#pragma unroll
            for (int w = 0; w < NWAVES; ++w)
#pragma unroll
                for (int s = 0; s < NSUB; ++s) c += hist[w][s][tid];
            scan[tid] = c;
            __syncthreads();

            // inclusive suffix scan: scan[t] = count(byte >= t)
            for (int off = 1; off < 256; off <<= 1) {
                unsigned v = scan[tid];
                unsigned a = (tid + off < 256) ? scan[tid + off] : 0u;
                __syncthreads();
                scan[tid] = v + a;
                __syncthreads();
            }

            unsigned ge = scan[tid];
            unsigned gt = (tid < 255) ? scan[tid + 1] : 0u;   // count(byte > t)
            if (ge >= remk && gt < remk) {                    // unique bucket holding the k-th
                s_bucket = (unsigned)tid;
                s_remk   = remk - gt;
            }
            __syncthreads();
            prefix |= (s_bucket << shift);
            pmask  |= (0xFFu << shift);
            remk    = s_remk;
            __syncthreads();
        }

        const unsigned thresh = prefix;     // exact k-th key (in flipped space)
        // sum of elements strictly above threshold (in flipped order)
        float local = 0.f;
#pragma unroll
        for (int e = 0; e < PER_TH; ++e) {
            unsigned key = f2k(vals[e]) ^ flip;
            if (key > thresh) local += vals[e];
        }
        // deterministic reduction: shuffle tree within wave32, serial across waves
#pragma unroll
        for (int off = 16; off > 0; off >>= 1) local += __shfl_down(local, off, 32);
        if (lane == 0) wsum[wave] = local;
        __syncthreads();

        if (tid == 0) {
            float tval = k2f(thresh ^ flip);       // threshold value in original space
            float s = (float)remk * tval;          // ties at threshold (all equal floats)
#pragma unroll
            for (int w = 0; w < NWAVES; ++w) s += wsum[w];
            if (side == 0) sum_top = s; else sum_bot = s;
        }
        __syncthreads();   // before next side reuses hist/scan/wsum
    }

    if (tid == 0)
        out[row] = sum_top * (1.0f / (float)KSEL) + ALPHA * (sum_bot * (1.0f / (float)KSEL));
}

extern "C" void kernel_launch(void* const* d_in, const int* in_sizes, int n_in,
                              void* d_out, int out_size, void* d_ws, size_t ws_size,
                              hipStream_t stream) {
    const float* x = (const float*)d_in[0];
    float* out = (float*)d_out;
    const int rows = in_sizes[0] / HW;   // 32 * 512 = 16384
    wildcat_pool_kernel<<<dim3(rows), dim3(THREADS), 0, stream>>>(x, out);
}